// LiquidEchoHead_75892072121048
// MI455X (gfx1250) — compile-verified
//
#include <hip/hip_runtime.h>
#include <math.h>

// ---------------------------------------------------------------------------
// LiquidEchoHead for MI455X (gfx1250).
// Bandwidth-bound (~768 MB compulsory traffic -> ~33us @ 23.3 TB/s); no GEMM
// structure so no WMMA. CDNA5 paths used: global_load_async_to_lds_b128 +
// s_wait_asynccnt for the LUT staging, non-temporal B128 loads/stores for the
// streamed 128MB arrays, wave32 shuffle reduction.
// ---------------------------------------------------------------------------

#define LUT_SIZE 4096
#define LUT_MASK (LUT_SIZE - 1)
#define BLOCK    256
#define DM       1024          // d_model fixed by the reference
#define TWO_PI   6.283185307179586

typedef float v4f __attribute__((ext_vector_type(4)));

// pos scale = LUT_SIZE / (2*pi), rounded to f32 exactly as jnp does
__device__ __forceinline__ float2 lut_sincos(float theta, const float2* lut) {
  float pos  = theta * 651.8986469044033f;
  float fl   = floorf(pos);
  float frac = pos - fl;                    // in [0,1)
  int   i0   = ((int)fl) & LUT_MASK;        // == jnp.mod(int32(floor), 4096)
  int   i1   = (i0 + 1) & LUT_MASK;
  float2 e0 = lut[i0];                      // ds_load_b64 gather
  float2 e1 = lut[i1];
  float om = 1.0f - frac;
  float2 r;
  r.x = e0.x * om + e1.x * frac;            // sin
  r.y = e0.y * om + e1.y * frac;            // cos
  return r;
}

// One-time table build in f64 (matches numpy float64 -> float32 table).
__global__ void lut_init_kernel(float2* __restrict__ lut) {
  int i = blockIdx.x * blockDim.x + threadIdx.x;
  if (i < LUT_SIZE) {
    double a = (double)i * (TWO_PI / (double)LUT_SIZE);
    lut[i] = make_float2((float)sin(a), (float)cos(a));
  }
}

template <bool USE_WS>
__global__ __launch_bounds__(BLOCK) void liquid_echo_kernel(
    const float* __restrict__ x_real, const float* __restrict__ x_imag,
    const float* __restrict__ t_vec,
    const float* __restrict__ mem_r,  const float* __restrict__ mem_i,
    const float* __restrict__ w_query, const float* __restrict__ b_query,
    const float* __restrict__ w_osc,   const float* __restrict__ b_osc,
    const float2* __restrict__ lut_g,
    float* __restrict__ out_er, float* __restrict__ out_ei,
    float* __restrict__ out_alpha, int n_rows)
{
  __shared__ float2 s_lut[LUT_SIZE];        // 32 KB interleaved (sin,cos)
  __shared__ float2 s_part[BLOCK / 32];     // 8 wave32 partials
  __shared__ float  s_alpha;

  const int tid = threadIdx.x;

  // ---- Stage LUT into LDS once per (persistent) block ----
  if (USE_WS) {
    // CDNA5 async global->LDS copy, tracked by ASYNCcnt.
    unsigned           lds_base = (unsigned)(size_t)(&s_lut[0]); // low 32 bits of
    unsigned long long gbase    = (unsigned long long)(const void*)lut_g; // generic = LDS offset
    #pragma unroll
    for (int it = 0; it < (LUT_SIZE * (int)sizeof(float2)) / (BLOCK * 16); ++it) {
      unsigned           loff  = lds_base + (unsigned)((tid + it * BLOCK) * 16);
      unsigned long long gaddr = gbase + (unsigned long long)((tid + it * BLOCK) * 16);
      asm volatile("global_load_async_to_lds_b128 %0, %1, off"
                   :: "v"(loff), "v"(gaddr) : "memory");
    }
    asm volatile("s_wait_asynccnt 0" ::: "memory");
  } else {
    // Fallback (no workspace): build table in-block with device sin/cos.
    const float step = (float)(TWO_PI / (double)LUT_SIZE);
    for (int i = tid; i < LUT_SIZE; i += BLOCK) {
      float a = (float)i * step;
      s_lut[i] = make_float2(sinf(a), cosf(a));
    }
  }
  __syncthreads();

  // ---- Per-d weights: each thread owns the same 4 d's for every row ----
  const v4f wq = ((const v4f*)w_query)[tid];
  const v4f bq = ((const v4f*)b_query)[tid];
  const v4f wo = ((const v4f*)w_osc)[tid];
  const v4f bo = ((const v4f*)b_osc)[tid];

  // ---- Persistent-block grid-stride loop over rows ----
  for (int row = blockIdx.x; row < n_rows; row += gridDim.x) {
    const size_t rbase = (size_t)row * DM;

    // Streamed once -> non-temporal loads (don't pollute L2).
    const v4f xr = __builtin_nontemporal_load((const v4f*)(x_real + rbase) + tid);
    const v4f mr = __builtin_nontemporal_load((const v4f*)(mem_r + rbase) + tid);
    const v4f mi = __builtin_nontemporal_load((const v4f*)(mem_i + rbase) + tid);

    // Phase 1: query phasor + interference partial sums
    float ir = 0.0f, ii = 0.0f;
    #pragma unroll
    for (int j = 0; j < 4; ++j) {
      float wl = 1.0f + fabsf(wq[j]);
      float th = xr[j] / wl + bq[j];
      float2 sc = lut_sincos(th, s_lut);      // (sin, cos)
      ir += sc.y * mr[j] + sc.x * mi[j];
      ii += sc.y * mi[j] - sc.x * mr[j];
    }
    // wave32 reduction
    #pragma unroll
    for (int off = 16; off > 0; off >>= 1) {
      ir += __shfl_down(ir, off, 32);
      ii += __shfl_down(ii, off, 32);
    }
    const int lane = tid & 31, wid = tid >> 5;
    if (lane == 0) s_part[wid] = make_float2(ir, ii);
    __syncthreads();
    if (tid == 0) {
      float sr = 0.0f, si = 0.0f;
      #pragma unroll
      for (int i = 0; i < BLOCK / 32; ++i) { sr += s_part[i].x; si += s_part[i].y; }
      float mag  = sqrtf(sr * sr + si * si);
      float z    = mag * (1.0f / 32.0f) - 2.0f;     // scale = sqrt(1024)
      float norm = 1.0f / (1.0f + expf(-z));        // sigmoid
      float a    = expf(-(1.0f - norm));            // k = 1
      s_alpha = a;
      out_alpha[row] = a;
    }
    __syncthreads();
    const float alpha   = s_alpha;
    const float one_m_a = 1.0f - alpha;

    // Phase 2: blend + oscillator evolution
    const v4f xi   = __builtin_nontemporal_load((const v4f*)(x_imag + rbase) + tid);
    const float tphi = t_vec[row] * 1.61803398874989484820f;   // t * PHI (f32)

    v4f er, ei;
    #pragma unroll
    for (int j = 0; j < 4; ++j) {
      float br  = alpha * xr[j] + one_m_a * mr[j];
      float bi  = alpha * xi[j] + one_m_a * mi[j];
      float wl  = 1.0f + fabsf(wo[j]);
      float thr = br / wl + bo[j] + tphi;
      float thi = bi / wl + bo[j] + tphi;
      float2 scr = lut_sincos(thr, s_lut);
      float2 sci = lut_sincos(thi, s_lut);
      er[j] = scr.y * sci.y - scr.x * sci.x;   // cos*cos - sin*sin
      ei[j] = scr.y * sci.x + scr.x * sci.y;   // cos*sin + sin*cos
    }
    __builtin_nontemporal_store(er, (v4f*)(out_er + rbase) + tid);
    __builtin_nontemporal_store(ei, (v4f*)(out_ei + rbase) + tid);
  }
}

extern "C" void kernel_launch(void* const* d_in, const int* in_sizes, int n_in,
                              void* d_out, int out_size, void* d_ws, size_t ws_size,
                              hipStream_t stream) {
  (void)n_in; (void)out_size;
  const float* x_real = (const float*)d_in[0];
  const float* x_imag = (const float*)d_in[1];
  const float* t_vec  = (const float*)d_in[2];
  const float* mem_r  = (const float*)d_in[3];
  const float* mem_i  = (const float*)d_in[4];
  const float* w_q    = (const float*)d_in[5];
  const float* b_q    = (const float*)d_in[6];
  const float* w_o    = (const float*)d_in[7];
  const float* b_o    = (const float*)d_in[8];

  const int B = in_sizes[2];     // rows (t has B elements)
  const int D = in_sizes[5];     // d_model (1024 per reference)

  float* out    = (float*)d_out;
  float* out_er = out;
  float* out_ei = out + (size_t)B * D;
  float* out_al = out + 2 * (size_t)B * D;

  int nblocks = B < 4096 ? B : 4096;   // persistent blocks, grid-stride rows

  const bool use_ws = (ws_size >= (size_t)LUT_SIZE * sizeof(float2));
  if (use_ws) {
    float2* lut = (float2*)d_ws;
    lut_init_kernel<<<LUT_SIZE / BLOCK, BLOCK, 0, stream>>>(lut);
    liquid_echo_kernel<true><<<nblocks, BLOCK, 0, stream>>>(
        x_real, x_imag, t_vec, mem_r, mem_i, w_q, b_q, w_o, b_o,
        (const float2*)lut, out_er, out_ei, out_al, B);
  } else {
    liquid_echo_kernel<false><<<nblocks, BLOCK, 0, stream>>>(
        x_real, x_imag, t_vec, mem_r, mem_i, w_q, b_q, w_o, b_o,
        nullptr, out_er, out_ei, out_al, B);
  }
}